// Project2DFeaturesCUDA_42597485641916
// MI455X (gfx1250) — compile-verified
//
#include <hip/hip_runtime.h>
#include <hip/hip_bf16.h>

// ---- problem constants (from reference) ----
#define B_      2
#define V_      4
#define H_      120
#define W_      160
#define C_      96
#define NVOX    200000
#define GRID_   128
#define NSTEPS  156
#define T0      2.0f      // DEPTH_MIN = 0.1/0.05
#define TSTEP   0.5f
#define NRAY    (B_*V_*H_*W_)          // 153600
#define OCC_INTS (B_*GRID_*GRID_*GRID_) // 4,194,304

typedef __attribute__((ext_vector_type(2))) float v2f;
typedef __attribute__((ext_vector_type(8))) float v8f;

#ifndef __has_builtin
#define __has_builtin(x) 0
#endif

// ---------------- Phase 1: per-batch min shift ----------------
__global__ void k_shift_init(int* __restrict__ shift) {
    if (threadIdx.x < 6) shift[threadIdx.x] = 0x7fffffff;
}

__global__ void k_shift_min(const int* __restrict__ coords, int* __restrict__ shift) {
    __shared__ int sm[6];
    if (threadIdx.x < 6) sm[threadIdx.x] = 0x7fffffff;
    __syncthreads();
    int i = blockIdx.x * blockDim.x + threadIdx.x;
    if (i < NVOX) {
        int b = coords[i*4 + 0];
        atomicMin(&sm[b*3 + 0], coords[i*4 + 1]);
        atomicMin(&sm[b*3 + 1], coords[i*4 + 2]);
        atomicMin(&sm[b*3 + 2], coords[i*4 + 3]);
    }
    __syncthreads();
    if (threadIdx.x < 6 && sm[threadIdx.x] != 0x7fffffff)
        atomicMin(&shift[threadIdx.x], sm[threadIdx.x]);
}

// ---------------- Phase 2: occupancy grid ----------------
__global__ void k_occ_init(int4* __restrict__ occ4) {
    int i = blockIdx.x * blockDim.x + threadIdx.x;   // exactly OCC_INTS/4 threads
    occ4[i] = make_int4(-1, -1, -1, -1);
}

__global__ void k_occ_scatter(const int* __restrict__ coords,
                              const int* __restrict__ shift,
                              int* __restrict__ occ) {
    int i = blockIdx.x * blockDim.x + threadIdx.x;
    if (i >= NVOX) return;
    int b = coords[i*4 + 0];
    int x = coords[i*4 + 1] - shift[b*3 + 0];
    int y = coords[i*4 + 2] - shift[b*3 + 1];
    int z = coords[i*4 + 3] - shift[b*3 + 2];
    // deterministic winner on duplicate coords (reference .set is unspecified)
    atomicMax(&occ[((b*GRID_ + z)*GRID_ + y)*GRID_ + x], i);
}

// ---------------- Phase 3: ray march (WMMA ray-direction transform) ----------------
__global__ void __launch_bounds__(256) k_raymarch(const float* __restrict__ vm,
                                                  const float* __restrict__ intr,
                                                  const int*   __restrict__ shift,
                                                  const int*   __restrict__ occ,
                                                  int*         __restrict__ tgt) {
    __shared__ float s_rd[256 * 4];
    const int lane    = threadIdx.x & 31;
    const int wave    = threadIdx.x >> 5;
    const int rayBase = blockIdx.x * 256 + wave * 32;   // wave handles 32 consecutive rays
    const int ray     = rayBase + lane;

    // (b, v, h, w0) are wave-uniform: W=160 and H*W divisible by 32
    const int b  = rayBase / (V_*H_*W_);
    const int v  = (rayBase / (H_*W_)) % V_;
    const int h  = (rayBase / W_) % H_;
    const int w0 = rayBase % W_;

    const float fx = intr[0], fy = intr[1], cx = intr[2], cy = intr[3];
    const float* M = vm + (b*V_ + v) * 16;              // row-major 4x4
    const float dyv = ((float)h + 0.5f - cy) / fy;

    const int col = lane & 15;
    const int hi  = lane >> 4;

#if __has_builtin(__builtin_amdgcn_wmma_f32_16x16x4_f32)
    // rd = R*d via V_WMMA_F32_16X16X4_F32:  A(16x4) = [dx dy 1 0] per ray, B(4x16) = R^T (cols>=3 zero)
    // B layout (wave32, 2 VGPRs): VGPR0 = rows {0,2} in lane halves, VGPR1 = rows {1,3}
    float r0 = 0.0f, r1 = 0.0f, r2 = 0.0f;
    if (col < 3) { r0 = M[col*4 + 0]; r1 = M[col*4 + 1]; r2 = M[col*4 + 2]; }
    v2f Bm;
    Bm[0] = hi ? r2 : r0;     // lanes<16: B[0][col]=R[col][0];  lanes>=16: B[2][col]=R[col][2]
    Bm[1] = hi ? 0.0f : r1;   // lanes<16: B[1][col]=R[col][1];  lanes>=16: B[3][col]=0
    v8f Cz = {0.f,0.f,0.f,0.f,0.f,0.f,0.f,0.f};
    #pragma unroll
    for (int g = 0; g < 2; ++g) {   // two 16-ray groups per wave
        // A layout (wave32, 2 VGPRs): lanes<16 carry K=0,1; lanes>=16 carry K=2,3
        v2f A;
        if (hi) { A[0] = 1.0f; A[1] = 0.0f; }
        else {
            float dxv = ((float)(w0 + g*16 + col) + 0.5f - cx) / fx;
            A[0] = dxv; A[1] = dyv;
        }
        v8f D = __builtin_amdgcn_wmma_f32_16x16x4_f32(false, A, false, Bm,
                                                      (short)0, Cz, false, false);
        // D layout: VGPR r -> row r (lanes 0-15, N=lane) / row r+8 (lanes 16-31, N=lane-16)
        if (col < 3) {
            int base = (wave*32 + g*16 + hi*8) * 4 + col;
            s_rd[base + 0*4] = D[0];
            s_rd[base + 1*4] = D[1];
            s_rd[base + 2*4] = D[2];
            s_rd[base + 3*4] = D[3];
            s_rd[base + 4*4] = D[4];
            s_rd[base + 5*4] = D[5];
            s_rd[base + 6*4] = D[6];
            s_rd[base + 7*4] = D[7];
        }
    }
#else
    {   // scalar fallback: rd = R * (dx, dy, 1)
        float dxv = ((float)(w0 + lane) + 0.5f - cx) / fx;
        s_rd[threadIdx.x*4 + 0] = M[0]*dxv + M[1]*dyv + M[2];
        s_rd[threadIdx.x*4 + 1] = M[4]*dxv + M[5]*dyv + M[6];
        s_rd[threadIdx.x*4 + 2] = M[8]*dxv + M[9]*dyv + M[10];
    }
#endif
    __syncthreads();

    const float rdx = s_rd[threadIdx.x*4 + 0];
    const float rdy = s_rd[threadIdx.x*4 + 1];
    const float rdz = s_rd[threadIdx.x*4 + 2];
    const float trx = M[3]  - (float)shift[b*3 + 0];
    const float tryy= M[7]  - (float)shift[b*3 + 1];
    const float trz = M[11] - (float)shift[b*3 + 2];

    const int* __restrict__ occb = occ + b * (GRID_*GRID_*GRID_);
    int t_ = NVOX;
    float tdep = T0;
    #pragma unroll 4
    for (int s = 0; s < NSTEPS; ++s, tdep += TSTEP) {
        float px = fmaf(tdep, rdx, trx);
        float py = fmaf(tdep, rdy, tryy);
        float pz = fmaf(tdep, rdz, trz);
        int ix = (int)floorf(px);
        int iy = (int)floorf(py);
        int iz = (int)floorf(pz);
        if (((unsigned)ix < (unsigned)GRID_) &
            ((unsigned)iy < (unsigned)GRID_) &
            ((unsigned)iz < (unsigned)GRID_)) {
            int id = occb[(iz*GRID_ + iy)*GRID_ + ix];
            if (id >= 0) { t_ = id; break; }     // first hit along the ray
        }
    }
    tgt[ray] = t_;
}

// ---------------- Phase 4: zero + scatter-accumulate ----------------
__global__ void k_zero(float4* __restrict__ p, int n4) {
    int i = blockIdx.x * blockDim.x + threadIdx.x;
    if (i < n4) p[i] = make_float4(0.f, 0.f, 0.f, 0.f);
}

__global__ void k_accum(const float* __restrict__ feats,
                        const int*   __restrict__ tgt,
                        float*       __restrict__ out) {
    int gid = blockIdx.x * blockDim.x + threadIdx.x;   // exactly NRAY*C_ threads
    int p = gid / C_;
    int c = gid - p * C_;
    int t = tgt[p];
    if (t < NVOX) {                     // t == NVOX -> dumped row, dropped (matches ref)
        atomicAdd(&out[t*C_ + c], feats[gid]);
        if (c == 0) atomicAdd(&out[NVOX*C_ + t], 1.0f);   // cnt region, exact int-in-f32
    }
}

// ---------------- Phase 5: normalize ----------------
__global__ void k_norm(float* __restrict__ out) {
    int gid = blockIdx.x * blockDim.x + threadIdx.x;   // exactly NVOX*C_ threads
    int i = gid / C_;
    float cnt = out[NVOX*C_ + i];
    out[gid] = out[gid] / (cnt + 1e-4f);
}

// ---------------- launcher ----------------
extern "C" void kernel_launch(void* const* d_in, const int* in_sizes, int n_in,
                              void* d_out, int out_size, void* d_ws, size_t ws_size,
                              hipStream_t stream) {
    const float* feats  = (const float*)d_in[0];   // (B,V,H,W,C) f32
    const int*   coords = (const int*)  d_in[1];   // (NVOX,4) i32
    const float* vm     = (const float*)d_in[2];   // (B,V,4,4) f32
    const float* intr   = (const float*)d_in[3];   // (4,) f32

    float* out = (float*)d_out;                    // [NVOX*C | NVOX cnt-as-f32]

    // workspace: tiny (robust to small ws_size): shift 6 ints @0, tgt @64B
    int* shift = (int*)d_ws;
    int* tgt   = (int*)((char*)d_ws + 64);
    // occupancy grid borrows d_out (16.8 MB << 77.6 MB); d_out is zeroed afterwards
    int* occ   = (int*)d_out;

    k_shift_init<<<1, 32, 0, stream>>>(shift);
    k_shift_min<<<(NVOX + 255)/256, 256, 0, stream>>>(coords, shift);
    k_occ_init<<<OCC_INTS/4/256, 256, 0, stream>>>((int4*)occ);
    k_occ_scatter<<<(NVOX + 255)/256, 256, 0, stream>>>(coords, shift, occ);
    k_raymarch<<<NRAY/256, 256, 0, stream>>>(vm, intr, shift, occ, tgt);

    const int totalF  = NVOX*C_ + NVOX;            // 19,400,000 (divisible by 4)
    k_zero<<<(totalF/4 + 255)/256, 256, 0, stream>>>((float4*)out, totalF/4);
    k_accum<<<(NRAY*C_)/256, 256, 0, stream>>>(feats, tgt, out);
    k_norm<<<(NVOX*C_)/256, 256, 0, stream>>>(out);
}